// DMXQuantizer_65386582114348
// MI455X (gfx1250) — compile-verified
//
#include <hip/hip_runtime.h>
#include <cstdint>

typedef float v4f __attribute__((ext_vector_type(4)));
typedef int   v4i __attribute__((ext_vector_type(4)));

#define BFP_AS1 __attribute__((address_space(1)))
#define BFP_AS3 __attribute__((address_space(3)))

// CDNA5 async global->LDS path (ASYNCcnt-tracked); proven present in round 1.
#if defined(__gfx1250__) &&                                                  \
    __has_builtin(__builtin_amdgcn_global_load_async_to_lds_b128) &&         \
    __has_builtin(__builtin_amdgcn_s_wait_asynccnt)
#define BFP_ASYNC 1
#else
#define BFP_ASYNC 0
#endif

#if defined(__gfx1250__) && __has_builtin(__builtin_amdgcn_ds_swizzle)
#define BFP_SWZ 1
#else
#define BFP_SWZ 0
#endif

// Butterfly max across lanes via ds_swizzle immediate pattern:
// group-of-32 mode, offset = xor_mask<<10 | or_mask<<5 | and_mask(0x1F).
template <int PAT>
__device__ __forceinline__ float bfp_swzmax(float v) {
#if BFP_SWZ
  const int p = __builtin_amdgcn_ds_swizzle(__float_as_int(v), PAT);
  return fmaxf(v, __int_as_float(p));
#else
  return fmaxf(v, __shfl_xor(v, (PAT >> 10) & 0x1F, 32));
#endif
}

// Clamp via v_med3_f32 when available (qmin < qmax always).
__device__ __forceinline__ float bfp_clamp(float r, float qmin, float qmax) {
#if defined(__gfx1250__) && __has_builtin(__builtin_amdgcn_fmed3f)
  return __builtin_amdgcn_fmed3f(r, qmin, qmax);
#else
  return fminf(fmaxf(r, qmin), qmax);
#endif
}

// Quantize one component given exact power-of-two scale/delta.
__device__ __forceinline__ float bfp_q(float xv, float scale, float delta,
                                       float qmin, float qmax) {
  float r = rintf(xv * scale);            // RNE, matches jnp.round
  return bfp_clamp(r, qmin, qmax) * delta;
}

__global__ __launch_bounds__(256) void bfp64_quant_kernel(
    const float* __restrict__ x, const int* __restrict__ bitsp,
    float* __restrict__ out, unsigned int nvec /* # of float4s, mult of 16 */) {
  const int bits = *bitsp;                              // uniform scalar load
  const int shift = bits - 2;                           // mantissa step exp
  const float qmax = (float)((1 << (bits - 1)) - 1);    // e.g. 127
  const float qmin = -(float)(1 << (bits - 1));         // e.g. -128

  const v4f* __restrict__ xv = (const v4f*)x;
  v4f* __restrict__ ov = (v4f*)out;

  const unsigned int tid = threadIdx.x;
  const unsigned int stride = gridDim.x * blockDim.x;
  const unsigned int idx0 = blockIdx.x * blockDim.x + tid;

#if BFP_ASYNC
  // Double-buffered LDS staging, one float4 slot per thread per buffer.
  // Each lane reads back only the 16B it asked for -> s_wait_asynccnt is the
  // only ordering needed (async loads complete in order within a wave).
  __shared__ v4f stage[2][256];
  if (idx0 < nvec) {
    __builtin_amdgcn_global_load_async_to_lds_b128(
        (BFP_AS1 v4i*)(v4f*)(xv + idx0),
        (BFP_AS3 v4i*)(v4f*)&stage[0][tid], 0, 0);
  }
  unsigned int parity = 0;
#endif

  for (unsigned int i = idx0; i < nvec; i += stride) {
    v4f v;
#if BFP_ASYNC
    const unsigned int nxt = i + stride;
    if (nxt < nvec) {
      __builtin_amdgcn_global_load_async_to_lds_b128(
          (BFP_AS1 v4i*)(v4f*)(xv + nxt),
          (BFP_AS3 v4i*)(v4f*)&stage[parity ^ 1][tid], 0, 0);
      __builtin_amdgcn_s_wait_asynccnt(1);   // wait for buffer `parity`
    } else {
      __builtin_amdgcn_s_wait_asynccnt(0);
    }
    v = stage[parity][tid];
    parity ^= 1;
#else
    if (i + stride < nvec)
      __builtin_prefetch((const void*)(xv + i + stride), 0, 1); // global_prefetch_b8
    v = __builtin_nontemporal_load(xv + i);
#endif

    // Per-lane max|.| over 4 elems, then 16-lane butterfly max: lanes 0..15
    // hold one 64-elem block, lanes 16..31 the next (xor masks < 16 never
    // cross the 16-lane group boundary). SWAPX1/2/4/8 ds_swizzle patterns.
    float ax = fmaxf(fmaxf(fabsf(v.x), fabsf(v.y)),
                     fmaxf(fabsf(v.z), fabsf(v.w)));
    ax = bfp_swzmax<0x041F>(ax);   // xor 1
    ax = bfp_swzmax<0x081F>(ax);   // xor 2
    ax = bfp_swzmax<0x101F>(ax);   // xor 4
    ax = bfp_swzmax<0x201F>(ax);   // xor 8

    v4f r;
    if (ax > 0.0f) {
      int e;
      (void)frexpf(ax, &e);                  // ax = m*2^e, m in [0.5,1)
      e -= 1;                                // exact floor(log2(ax))
      const float delta = ldexpf(1.0f, e - shift);   // exact 2^(e-shift)
      const float scale = ldexpf(1.0f, shift - e);   // exact 1/delta
      r.x = bfp_q(v.x, scale, delta, qmin, qmax);
      r.y = bfp_q(v.y, scale, delta, qmin, qmax);
      r.z = bfp_q(v.z, scale, delta, qmin, qmax);
      r.w = bfp_q(v.w, scale, delta, qmin, qmax);
    } else {
      r = 0.0f;                              // zero block
    }

    __builtin_nontemporal_store(r, ov + i);
  }
}

extern "C" void kernel_launch(void* const* d_in, const int* in_sizes, int n_in,
                              void* d_out, int out_size, void* d_ws, size_t ws_size,
                              hipStream_t stream) {
  const float* x = (const float*)d_in[0];
  const int* bits = (const int*)d_in[1];
  float* out = (float*)d_out;

  const unsigned long long n = (unsigned long long)in_sizes[0]; // 4096*8192
  const unsigned int nvec = (unsigned int)(n / 4);              // float4 count

  const int threads = 256;                      // 8 wave32s / workgroup
  unsigned long long blocks = (nvec + threads - 1) / threads;
  if (blocks > 8192ull) blocks = 8192ull;       // grid-stride, ~4 iters/thread

  bfp64_quant_kernel<<<(unsigned)blocks, threads, 0, stream>>>(x, bits, out, nvec);
}